// VanillaLSTM_46557445488843
// MI455X (gfx1250) — compile-verified
//
#include <hip/hip_runtime.h>
#include <hip/hip_bf16.h>

typedef float v2f __attribute__((ext_vector_type(2)));
typedef float v8f __attribute__((ext_vector_type(8)));

#define NSQ   3
#define BSZ   64
#define TSZ   128
#define HSZ   96
#define BLK   32
#define VSZ   1024            // BLK*BLK
#define NCELL (BSZ * NSQ * NSQ)   // 576
#define GSTRIDE (NSQ * NSQ * 4)   // 36 floats per sample in gates buffer
#define NSAMP (TSZ * BSZ)         // 8192 (t,b) samples

// ---------------------------------------------------------------------------
// Kernel 1: x_gates[s][ij][g] = sum_v xs[s][ij][v] * W_ih[ij][g][v]
// One wave (32 threads) per (16-sample tile, ij block).
// Uses V_WMMA_F32_16X16X4_F32 (full fp32 precision) accumulating over K=1024
// in chunks of 4, with 4 independent accumulator chains for ILP.
// ---------------------------------------------------------------------------
__global__ __launch_bounds__(32)
void lstm_gates_wmma(const float* __restrict__ x,
                     const float* __restrict__ W_ih,
                     float* __restrict__ gates) {
    const int lane = threadIdx.x;        // 0..31, wave32
    const int m    = lane & 15;          // A-matrix row within tile / B column n
    const int kh   = lane >> 4;          // 0: K={0,1}, 1: K={2,3} within chunk
    const int tile = blockIdx.x;         // 0..511  (sample tile of 16)
    const int ij   = blockIdx.y;         // 0..8    (spatial block)

    // Sample index -> (t, b).  16 consecutive samples share t (64 % 16 == 0).
    const int s = tile * 16 + m;
    const int t = s >> 6;
    const int b = s & 63;
    const int bi = ij / 3, bj = ij % 3;

    // Base of this sample's 32x32 block inside x[b, 0, t, :, :]
    const float* __restrict__ xb =
        x + ((size_t)(b * 2) * TSZ + t) * (HSZ * HSZ)
          + (size_t)(bi * BLK) * HSZ + (size_t)(bj * BLK);

    // B-matrix (4x16, K x gate): lane provides column n = m.  Lanes n>=4 are
    // zero-padding: clamp the row pointer in-bounds and multiply by 0 so EXEC
    // stays uniform (WMMA requires EXEC == all ones).
    const int   n     = m;
    const float wmask = (n < 4) ? 1.0f : 0.0f;
    const int   wrow  = (n < 4) ? n : 3;
    const float* __restrict__ wb = W_ih + (size_t)(ij * 4 + wrow) * VSZ;

    v8f acc[4] = {v8f{}, v8f{}, v8f{}, v8f{}};

    for (int k0 = 0; k0 < VSZ; k0 += 16) {
#pragma unroll
        for (int c = 0; c < 4; ++c) {
            const int v = k0 + 4 * c + 2 * kh;           // even -> pair in one row
            // A fragment: xs[sample m][v], xs[sample m][v+1]
            v2f a = *(const v2f*)(xb + (v >> 5) * HSZ + (v & 31));
            // B fragment: W_ih[ij][n][v], W_ih[ij][n][v+1]  (zeroed for n>=4)
            v2f w = *(const v2f*)(wb + v);
            w.x *= wmask;
            w.y *= wmask;
            // D = A(16x4) * B(4x16) + C  — fp32 WMMA
            acc[c] = __builtin_amdgcn_wmma_f32_16x16x4_f32(
                false, a, false, w, (short)0, acc[c], false, false);
        }
    }

    v8f accs = (acc[0] + acc[1]) + (acc[2] + acc[3]);

    // C/D layout: VGPR p holds (M = p + 8*kh, N = lane & 15).  Valid gates: N<4.
    if (n < 4) {
#pragma unroll
        for (int p = 0; p < 8; ++p) {
            const int sm = tile * 16 + p + 8 * kh;       // sample for this VGPR
            gates[(size_t)sm * GSTRIDE + ij * 4 + n] = accs[p];
        }
    }
}

// ---------------------------------------------------------------------------
// Kernel 2: sequential LSTM scan, hidden_size = 1 per (b, ij) cell.
// 576 independent cells; gates buffer (1.18 MB) is L2-resident.
// ---------------------------------------------------------------------------
__device__ __forceinline__ float sigmoidf_(float v) {
    return 1.0f / (1.0f + __expf(-v));
}

__global__ __launch_bounds__(64)
void lstm_scan(const float* __restrict__ gates,
               const float* __restrict__ W_hh,
               const float* __restrict__ b_ih,
               const float* __restrict__ b_hh,
               float* __restrict__ out) {
    const int cell = blockIdx.x * blockDim.x + threadIdx.x;   // 0..575
    if (cell >= NCELL) return;
    const int b  = cell / (NSQ * NSQ);
    const int ij = cell % (NSQ * NSQ);

    const float wh0 = W_hh[ij * 4 + 0], wh1 = W_hh[ij * 4 + 1];
    const float wh2 = W_hh[ij * 4 + 2], wh3 = W_hh[ij * 4 + 3];
    const float bb0 = b_ih[ij * 4 + 0] + b_hh[ij * 4 + 0];
    const float bb1 = b_ih[ij * 4 + 1] + b_hh[ij * 4 + 1];
    const float bb2 = b_ih[ij * 4 + 2] + b_hh[ij * 4 + 2];
    const float bb3 = b_ih[ij * 4 + 3] + b_hh[ij * 4 + 3];

    float h = 0.0f, cst = 0.0f;
    for (int t = 0; t < TSZ; ++t) {
        const float* g = gates + ((size_t)(t * BSZ + b) * GSTRIDE + ij * 4);
        const float gi = g[0] + bb0 + wh0 * h;
        const float gf = g[1] + bb1 + wh1 * h;
        const float gg = g[2] + bb2 + wh2 * h;
        const float go = g[3] + bb3 + wh3 * h;
        const float i_ = sigmoidf_(gi);
        const float f_ = sigmoidf_(gf);
        const float g_ = tanhf(gg);
        const float o_ = sigmoidf_(go);
        cst = f_ * cst + i_ * g_;
        h   = o_ * tanhf(cst);
        out[(size_t)t * NCELL + cell] = h;   // (t, b, i, j) with cell = b*9 + ij
    }
}

// ---------------------------------------------------------------------------
extern "C" void kernel_launch(void* const* d_in, const int* in_sizes, int n_in,
                              void* d_out, int out_size, void* d_ws, size_t ws_size,
                              hipStream_t stream) {
    const float* x    = (const float*)d_in[0];
    const float* W_ih = (const float*)d_in[1];
    const float* W_hh = (const float*)d_in[2];
    const float* b_ih = (const float*)d_in[3];
    const float* b_hh = (const float*)d_in[4];
    float* out   = (float*)d_out;
    float* gates = (float*)d_ws;   // NSAMP * GSTRIDE floats = 1.18 MB

    dim3 grid1(NSAMP / 16, NSQ * NSQ);   // 512 x 9 waves
    lstm_gates_wmma<<<grid1, 32, 0, stream>>>(x, W_ih, gates);

    lstm_scan<<<(NCELL + 63) / 64, 64, 0, stream>>>(gates, W_hh, b_ih, b_hh, out);
}